// SGC_63677185130716
// MI455X (gfx1250) — compile-verified
//
#include <hip/hip_runtime.h>
#include <math.h>

#define N_NODES   100000
#define N_FEAT    128
#define N_CLASSES 40

typedef float v2f __attribute__((ext_vector_type(2)));
typedef float v8f __attribute__((ext_vector_type(8)));

// ---------------- degree / norm ----------------

__global__ void sgc_deg_init(float* __restrict__ deg, int n) {
  int i = blockIdx.x * blockDim.x + threadIdx.x;
  if (i < n) deg[i] = 1.0f;  // self-loop contributes 1
}

__global__ void sgc_deg_scatter(const int* __restrict__ dst, float* __restrict__ deg, int e) {
  int i = blockIdx.x * blockDim.x + threadIdx.x;
  if (i < e) atomicAdd(&deg[dst[i]], 1.0f);
}

__global__ void sgc_dinv(float* __restrict__ deg, int n) {
  int i = blockIdx.x * blockDim.x + threadIdx.x;
  if (i < n) deg[i] = rsqrtf(deg[i]);  // deg >= 1 always (self loop)
}

// ---------------- propagation: h_out = Ahat @ h_in ----------------

// self-loop term + init: h_out[i] = dinv[i]^2 * h_in[i]
__global__ void sgc_selfloop(const float* __restrict__ hin, const float* __restrict__ dinv,
                             float* __restrict__ hout, int n) {
  int i = blockIdx.x * blockDim.x + threadIdx.x;  // one float4 per thread
  int total = n * (N_FEAT / 4);
  if (i < total) {
    int node = i >> 5;  // 32 float4s per node
    float s = dinv[node];
    s = s * s;
    float4 v = ((const float4*)hin)[i];
    v.x *= s; v.y *= s; v.z *= s; v.w *= s;
    ((float4*)hout)[i] = v;
  }
}

// edge scatter: one wave of 32 lanes per edge, 4 features per lane
__global__ void sgc_edge_spmm(const float* __restrict__ hin, const int* __restrict__ src,
                              const int* __restrict__ dst, const float* __restrict__ dinv,
                              float* __restrict__ hout, int e) {
  int gid  = blockIdx.x * blockDim.x + threadIdx.x;
  int edge = gid >> 5;
  int lane = gid & 31;
  if (edge < e) {
    int s = src[edge];
    int d = dst[edge];
    float norm = dinv[s] * dinv[d];
    float4 v = ((const float4*)(hin + (size_t)s * N_FEAT))[lane];
    float* outp = hout + (size_t)d * N_FEAT + lane * 4;
    atomicAdd(outp + 0, v.x * norm);
    atomicAdd(outp + 1, v.y * norm);
    atomicAdd(outp + 2, v.z * norm);
    atomicAdd(outp + 3, v.w * norm);
  }
}

// ---------------- WMMA GEMM: logits = h @ W^T + bias ----------------
// Block = 96 threads = 3 waves. Block handles 16 rows (M tile), wave w handles
// columns [16w, 16w+16). K = 128 consumed in steps of 4 with
// V_WMMA_F32_16X16X4_F32 (fp32 A/B, fp32 accumulate).

__global__ __launch_bounds__(96) void sgc_gemm_wmma(
    const float* __restrict__ h, const float* __restrict__ w,
    const float* __restrict__ bias, float* __restrict__ out) {
  __shared__ float h_lds[16 * N_FEAT];   // 8 KB
  __shared__ float w_lds[48 * N_FEAT];   // 24 KB, rows 40..47 zero-padded

  const int tid = threadIdx.x;
  const int m0  = blockIdx.x * 16;       // 6250 * 16 == 100000 exactly

  for (int idx = tid; idx < 16 * N_FEAT; idx += 96)
    h_lds[idx] = h[(size_t)m0 * N_FEAT + idx];
  for (int idx = tid; idx < 48 * N_FEAT; idx += 96) {
    int r = idx >> 7;  // /128
    w_lds[idx] = (r < N_CLASSES) ? w[idx] : 0.0f;
  }
  __syncthreads();

  const int wave    = tid >> 5;
  const int lane    = tid & 31;
  const int laneLow = lane & 15;
  const int kHalf   = (lane >> 4) << 1;  // lanes 16-31 hold K+2,K+3
  const int n0      = wave << 4;

  const float* arow = &h_lds[laneLow * N_FEAT];
  const float* brow = &w_lds[(n0 + laneLow) * N_FEAT];

  v8f acc = {};
#pragma unroll
  for (int k = 0; k < N_FEAT; k += 4) {
    v2f a, b;
    a.x = arow[k + kHalf];
    a.y = arow[k + kHalf + 1];
    b.x = brow[k + kHalf];
    b.y = brow[k + kHalf + 1];
    // (neg_a, A, neg_b, B, c_mod, C, reuse_a, reuse_b)
    acc = __builtin_amdgcn_wmma_f32_16x16x4_f32(false, a, false, b, (short)0,
                                                acc, false, false);
  }

  // C/D layout: VGPR v -> row m0 + v + 8*(lane>>4), col n0 + (lane&15)
  const int col = n0 + laneLow;
  if (col < N_CLASSES) {
    const float bv = bias[col];
    const int rbase = m0 + ((lane >> 4) << 3);
#pragma unroll
    for (int v = 0; v < 8; ++v)
      out[(size_t)(rbase + v) * N_CLASSES + col] = acc[v] + bv;
  }
}

// ---------------- log_softmax (in-place, one thread per row) ----------------

__global__ void sgc_logsoftmax(float* __restrict__ out, int n) {
  int i = blockIdx.x * blockDim.x + threadIdx.x;
  if (i < n) {
    float* row = out + (size_t)i * N_CLASSES;
    float vals[N_CLASSES];
    float m = -INFINITY;
#pragma unroll
    for (int c = 0; c < N_CLASSES; ++c) {
      vals[c] = row[c];
      m = fmaxf(m, vals[c]);
    }
    float s = 0.0f;
#pragma unroll
    for (int c = 0; c < N_CLASSES; ++c) s += __expf(vals[c] - m);
    float lse = m + __logf(s);
#pragma unroll
    for (int c = 0; c < N_CLASSES; ++c) row[c] = vals[c] - lse;
  }
}

// ---------------- launch ----------------

extern "C" void kernel_launch(void* const* d_in, const int* in_sizes, int n_in,
                              void* d_out, int out_size, void* d_ws, size_t ws_size,
                              hipStream_t stream) {
  const float* x    = (const float*)d_in[0];
  const float* w    = (const float*)d_in[1];
  const float* bias = (const float*)d_in[2];
  const int*   ei   = (const int*)d_in[3];

  const int n = N_NODES;
  const int E = in_sizes[3] / 2;
  const int* src = ei;       // edge_index[0]
  const int* dst = ei + E;   // edge_index[1]

  char* ws = (char*)d_ws;
  const size_t H_BYTES = (size_t)n * N_FEAT * sizeof(float);  // 51.2 MB
  float* dinv = (float*)ws;                                   // 400 KB (as deg, then dinv)
  float* h1   = (float*)(ws + 512 * 1024);
  float* h2   = (float*)(ws + 512 * 1024 + ((H_BYTES + 511) & ~(size_t)511));

  float* out = (float*)d_out;

  const int T = 256;
  // 1) degree (unit weights scattered onto targets, + self loops)
  sgc_deg_init<<<(n + T - 1) / T, T, 0, stream>>>(dinv, n);
  sgc_deg_scatter<<<(E + T - 1) / T, T, 0, stream>>>(dst, dinv, E);
  sgc_dinv<<<(n + T - 1) / T, T, 0, stream>>>(dinv, n);

  // 2) hop 1: h1 = Ahat @ x
  const int elem4 = n * (N_FEAT / 4);
  sgc_selfloop<<<(elem4 + T - 1) / T, T, 0, stream>>>(x, dinv, h1, n);
  sgc_edge_spmm<<<((size_t)E * 32 + T - 1) / T, T, 0, stream>>>(x, src, dst, dinv, h1, E);

  // 3) hop 2: h2 = Ahat @ h1
  sgc_selfloop<<<(elem4 + T - 1) / T, T, 0, stream>>>(h1, dinv, h2, n);
  sgc_edge_spmm<<<((size_t)E * 32 + T - 1) / T, T, 0, stream>>>(h1, src, dst, dinv, h2, E);

  // 4) logits = h2 @ W^T + bias  (WMMA f32 16x16x4)
  sgc_gemm_wmma<<<n / 16, 96, 0, stream>>>(h2, w, bias, out);

  // 5) log_softmax in place
  sgc_logsoftmax<<<(n + T - 1) / T, T, 0, stream>>>(out, n);
}